// SphereConv_37237366456772
// MI455X (gfx1250) — compile-verified
//
#include <hip/hip_runtime.h>

typedef __attribute__((ext_vector_type(2))) float v2f;
typedef __attribute__((ext_vector_type(8))) float v8f;

#define NB   16
#define NP   2048
#define KNN  16
#define DD   4
#define CPF  64
#define ADDC 64
#define CCH  256
#define CIN  128
#define HH   32
#define PTS_ELEMS (NB * NP * 3) /* 98304 */
#define INV_BN 0.99999500003749978f /* 1/sqrt(1+1e-5) */

static __device__ __forceinline__ v8f wmma4(v2f a, v2f b, v8f c) {
  // V_WMMA_F32_16X16X4_F32 : D = A(16x4) * B(4x16) + C(16x16), all f32
  return __builtin_amdgcn_wmma_f32_16x16x4_f32(false, a, false, b, (short)0, c,
                                               false, false);
}

// ---------------------------------------------------------------------------
// Kernel 1: kNN top-16 (ascending distance; order among equal distances not guaranteed)
// one thread per (n,p); candidate points tiled through LDS
// ---------------------------------------------------------------------------
__global__ __launch_bounds__(256) void knn_kernel(const float* __restrict__ pts,
                                                  int* __restrict__ idx_out) {
  __shared__ float sx[256], sy[256], sz[256];
  const int t = blockIdx.x * 256 + threadIdx.x; // 0..32767, 8 blocks per batch
  const int n = t >> 11;
  const int p = t & (NP - 1);
  const float* pb = pts + (size_t)n * NP * 3;
  const float cx = pb[p * 3 + 0];
  const float cy = pb[p * 3 + 1];
  const float cz = pb[p * 3 + 2];

  float bd[KNN];
  int bi[KNN];
#pragma unroll
  for (int j = 0; j < KNN; ++j) {
    bd[j] = 3.4e38f;
    bi[j] = 0;
  }

  for (int q0 = 0; q0 < NP; q0 += 256) {
    const int ql = q0 + threadIdx.x;
    sx[threadIdx.x] = pb[ql * 3 + 0];
    sy[threadIdx.x] = pb[ql * 3 + 1];
    sz[threadIdx.x] = pb[ql * 3 + 2];
    __syncthreads();
    for (int qq = 0; qq < 256; ++qq) {
      const float dx = sx[qq] - cx;
      const float dy = sy[qq] - cy;
      const float dz = sz[qq] - cz;
      const float d = dx * dx + dy * dy + dz * dz;
      if (d < bd[KNN - 1]) { // rare path: swap-based insertion; equal distances may reorder
        float cd = d;
        int ci = q0 + qq;
#pragma unroll
        for (int j = 0; j < KNN; ++j) {
          if (cd < bd[j]) {
            const float td = bd[j]; bd[j] = cd; cd = td;
            const int ti = bi[j];   bi[j] = ci; ci = ti;
          }
        }
      }
    }
    __syncthreads();
  }
#pragma unroll
  for (int j = 0; j < KNN; ++j) idx_out[(size_t)t * KNN + j] = bi[j];
}

// ---------------------------------------------------------------------------
// Kernel 2: per-point MLP (fc1->bn1->relu scalar, fc2->bn2->relu via WMMA),
// fused concat + windowed max-pool; one wave per (n,p)
// writes x[n][g][p][c*4+d]  (g=ch>>5, c=ch&31; ch 0..63 = h feats, 64..127 = fts)
// ---------------------------------------------------------------------------
__global__ __launch_bounds__(256) void mlp_pool_kernel(
    const float* __restrict__ pts, const float* __restrict__ fts_prev,
    const float* __restrict__ fc1_w, const float* __restrict__ fc1_b,
    const float* __restrict__ fc2_w, const float* __restrict__ fc2_b,
    const float* __restrict__ bn1_g, const float* __restrict__ bn1_b,
    const float* __restrict__ bn2_g, const float* __restrict__ bn2_b,
    const int* __restrict__ idx_ws, float* __restrict__ xw) {
  __shared__ __align__(16) float s_fc2w[CPF * HH];  // 64 x 32
  __shared__ __align__(16) float s_h1[8][KNN * HH]; // per-wave 16 x 32
  __shared__ int s_idx[8][KNN];

  const int lane = threadIdx.x & 31;
  const int wv = threadIdx.x >> 5;
  const int pt = blockIdx.x * 8 + wv;
  const int n = pt >> 11;
  const int p = pt & (NP - 1);

  for (int i = threadIdx.x; i < CPF * HH; i += 256) s_fc2w[i] = fc2_w[i];
  if (lane < KNN) s_idx[wv][lane] = idx_ws[(size_t)pt * KNN + lane];
  __syncthreads();

  const int m = lane & 15; // neighbor row / tile column
  const int hi = lane >> 4;

  // fc1 + bn1 + relu: lane handles neighbor m, feature range [hi*16, hi*16+16)
  {
    const float* pbase = pts + (size_t)n * NP * 3;
    const int im = s_idx[wv][m];
    const float rx = pbase[im * 3 + 0] - pbase[p * 3 + 0];
    const float ry = pbase[im * 3 + 1] - pbase[p * 3 + 1];
    const float rz = pbase[im * 3 + 2] - pbase[p * 3 + 2];
    const int f0 = hi * 16;
#pragma unroll
    for (int f = 0; f < 16; ++f) {
      const int fe = f0 + f;
      float v = rx * fc1_w[fe * 3 + 0] + ry * fc1_w[fe * 3 + 1] +
                rz * fc1_w[fe * 3 + 2] + fc1_b[fe];
      v = fmaxf(v * (bn1_g[fe] * INV_BN) + bn1_b[fe], 0.0f);
      s_h1[wv][m * HH + fe] = v;
    }
  }
  __syncthreads();

  // fc2 via WMMA: M=16 neighbors, N=64 features (4 tiles), K=32 (8 steps of 4)
  v8f acc[4];
#pragma unroll
  for (int t4 = 0; t4 < 4; ++t4) acc[t4] = (v8f){0, 0, 0, 0, 0, 0, 0, 0};
#pragma unroll
  for (int ks = 0; ks < HH / 4; ++ks) {
    const int k0 = ks * 4 + 2 * hi;
    const v2f a = *(const v2f*)&s_h1[wv][m * HH + k0];
#pragma unroll
    for (int t4 = 0; t4 < 4; ++t4) {
      const v2f b = *(const v2f*)&s_fc2w[(t4 * 16 + m) * HH + k0];
      acc[t4] = wmma4(a, b, acc[t4]);
    }
  }

  // bias + bn2 + relu + max-pool over neighbor windows of 4, directly on C/D
  // layout: vgpr j -> neighbor j + 8*hi ; lane's column -> feature t4*16+m
#pragma unroll
  for (int t4 = 0; t4 < 4; ++t4) {
    const int feat = t4 * 16 + m;
    const float sc = bn2_g[feat] * INV_BN;
    const float sh = bn2_b[feat];
    const float bb = fc2_b[feat];
    float v[8];
#pragma unroll
    for (int j = 0; j < 8; ++j)
      v[j] = fmaxf((acc[t4][j] + bb) * sc + sh, 0.0f);
    const float p0v = fmaxf(fmaxf(v[0], v[1]), fmaxf(v[2], v[3]));
    const float p1v = fmaxf(fmaxf(v[4], v[5]), fmaxf(v[6], v[7]));
    const int g = feat >> 5;
    const int c = feat & 31;
    float* base = xw + (((size_t)(n * DD + g) * NP + p) * CIN + c * DD);
    base[2 * hi + 0] = p0v; // windows 0/1 (lanes<16) or 2/3 (lanes>=16)
    base[2 * hi + 1] = p1v;
  }

  // gathered fts_prev max-pool: each lane handles channels lane and lane+32
  for (int cc = lane; cc < ADDC; cc += 32) {
    const float* fb = fts_prev + ((size_t)n * ADDC + cc) * NP;
    float mw0 = -3.4e38f, mw1 = -3.4e38f, mw2 = -3.4e38f, mw3 = -3.4e38f;
#pragma unroll
    for (int k = 0; k < KNN; ++k) {
      const float val = fb[s_idx[wv][k]];
      if (k < 4) mw0 = fmaxf(mw0, val);
      else if (k < 8) mw1 = fmaxf(mw1, val);
      else if (k < 12) mw2 = fmaxf(mw2, val);
      else mw3 = fmaxf(mw3, val);
    }
    const int ch = ADDC + cc;
    const int g = ch >> 5;
    const int c2 = ch & 31;
    float* base = xw + (((size_t)(n * DD + g) * NP + p) * CIN + c2 * DD);
    base[0] = mw0; base[1] = mw1; base[2] = mw2; base[3] = mw3;
  }
}

// ---------------------------------------------------------------------------
// Kernel 3: grouped conv1 as per-group GEMM Y[g] = W[g](64x128) @ X[g]^T(128xP)
// one wave per 16(o) x 16(p) tile, K=128 -> 32 WMMAs; fused bias+bn3+relu
// ---------------------------------------------------------------------------
__global__ __launch_bounds__(256) void conv1_kernel(
    const float* __restrict__ xw, const float* __restrict__ w1,
    const float* __restrict__ b1, const float* __restrict__ bn3_g,
    const float* __restrict__ bn3_b, float* __restrict__ y_ws) {
  const int lane = threadIdx.x & 31;
  const int wid = (blockIdx.x * 256 + threadIdx.x) >> 5;
  const int ptile = wid & 127;
  int r = wid >> 7;
  const int ot = r & 3; r >>= 2;
  const int g = r & 3;
  const int n = r >> 2;
  const int m = lane & 15;
  const int hi = lane >> 4;

  const float* arow = w1 + (size_t)(g * 64 + ot * 16 + m) * CIN; // [o][c*4+d]
  const float* brow = xw + ((size_t)(n * DD + g) * NP + ptile * 16 + m) * CIN;

  v8f acc = (v8f){0, 0, 0, 0, 0, 0, 0, 0};
#pragma unroll
  for (int ks = 0; ks < CIN / 4; ++ks) {
    const int k0 = ks * 4 + 2 * hi;
    const v2f a = *(const v2f*)(arow + k0);
    const v2f b = *(const v2f*)(brow + k0);
    acc = wmma4(a, b, acc);
  }
  const int p = ptile * 16 + m;
#pragma unroll
  for (int j = 0; j < 8; ++j) {
    const int o = g * 64 + ot * 16 + j + 8 * hi;
    const float val =
        fmaxf((acc[j] + b1[o]) * (bn3_g[o] * INV_BN) + bn3_b[o], 0.0f);
    y_ws[((size_t)n * CCH + o) * NP + p] = val;
  }
}

// ---------------------------------------------------------------------------
// Kernel 4: conv2 (1x1): Z = W2(64x256) @ Y(256xP), writes into output concat
// ---------------------------------------------------------------------------
__global__ __launch_bounds__(256) void conv2_kernel(
    const float* __restrict__ y_ws, const float* __restrict__ w2,
    const float* __restrict__ b2, float* __restrict__ out) {
  const int lane = threadIdx.x & 31;
  const int wid = (blockIdx.x * 256 + threadIdx.x) >> 5;
  const int ptile = wid & 127;
  int r = wid >> 7;
  const int ot = r & 3;
  const int n = r >> 2;
  const int m = lane & 15;
  const int hi = lane >> 4;

  const float* arow = w2 + (size_t)(ot * 16 + m) * CCH;
  const float* bcol = y_ws + (size_t)n * CCH * NP + ptile * 16 + m;

  v8f acc = (v8f){0, 0, 0, 0, 0, 0, 0, 0};
#pragma unroll 8
  for (int ks = 0; ks < CCH / 4; ++ks) {
    const int k0 = ks * 4 + 2 * hi;
    const v2f a = *(const v2f*)(arow + k0);
    v2f b;
    b.x = bcol[(size_t)k0 * NP];
    b.y = bcol[(size_t)(k0 + 1) * NP];
    acc = wmma4(a, b, acc);
  }
  const int p = ptile * 16 + m;
#pragma unroll
  for (int j = 0; j < 8; ++j) {
    const int o = ot * 16 + j + 8 * hi;
    out[(size_t)PTS_ELEMS + ((size_t)n * CIN + ADDC + o) * NP + p] =
        acc[j] + b2[o];
  }
}

// ---------------------------------------------------------------------------
// Kernel 5: place fts_prev into the first 64 channels of the output concat
// ---------------------------------------------------------------------------
__global__ __launch_bounds__(256) void concat_copy_kernel(
    const float* __restrict__ fts_prev, float* __restrict__ out) {
  const int t = blockIdx.x * 256 + threadIdx.x; // < 16*64*2048
  const int n = t >> 17;
  const int rem = t & (ADDC * NP - 1);
  out[(size_t)PTS_ELEMS + (size_t)n * CIN * NP + rem] = fts_prev[t];
}

extern "C" void kernel_launch(void* const* d_in, const int* in_sizes, int n_in,
                              void* d_out, int out_size, void* d_ws,
                              size_t ws_size, hipStream_t stream) {
  (void)in_sizes; (void)n_in; (void)out_size; (void)ws_size;
  const float* pts = (const float*)d_in[0];
  const float* fts_prev = (const float*)d_in[1];
  const float* fc1_w = (const float*)d_in[2];
  const float* fc1_b = (const float*)d_in[3];
  const float* fc2_w = (const float*)d_in[4];
  const float* fc2_b = (const float*)d_in[5];
  const float* bn1_g = (const float*)d_in[6];
  const float* bn1_b = (const float*)d_in[7];
  const float* bn2_g = (const float*)d_in[8];
  const float* bn2_b = (const float*)d_in[9];
  const float* bn3_g = (const float*)d_in[10];
  const float* bn3_b = (const float*)d_in[11];
  const float* conv1_w = (const float*)d_in[12];
  const float* conv1_b = (const float*)d_in[13];
  const float* conv2_w = (const float*)d_in[14];
  const float* conv2_b = (const float*)d_in[15];
  float* out = (float*)d_out;

  char* ws = (char*)d_ws;
  int* idx_ws = (int*)ws;                                   // 2 MB
  float* xw = (float*)(ws + ((size_t)2 << 20));             // 64 MB
  float* y_ws = (float*)(ws + ((size_t)66 << 20));          // 32 MB

  knn_kernel<<<NB * NP / 256, 256, 0, stream>>>(pts, idx_ws);
  mlp_pool_kernel<<<NB * NP / 8, 256, 0, stream>>>(
      pts, fts_prev, fc1_w, fc1_b, fc2_w, fc2_b, bn1_g, bn1_b, bn2_g, bn2_b,
      idx_ws, xw);
  conv1_kernel<<<NB * DD * 4 * 128 / 8, 256, 0, stream>>>(xw, conv1_w, conv1_b,
                                                          bn3_g, bn3_b, y_ws);
  conv2_kernel<<<NB * 4 * 128 / 8, 256, 0, stream>>>(y_ws, conv2_w, conv2_b,
                                                     out);
  concat_copy_kernel<<<NB * ADDC * NP / 256, 256, 0, stream>>>(fts_prev, out);
  hipMemcpyAsync(d_out, d_in[0], (size_t)PTS_ELEMS * sizeof(float),
                 hipMemcpyDeviceToDevice, stream);
}